// PawletteMambaBlock_29566554865935
// MI455X (gfx1250) — compile-verified
//
#include <hip/hip_runtime.h>
#include <hip/hip_bf16.h>
#include <math.h>

#define D_MODEL   672
#define D_STATE   128
#define D_CONV    4
#define HEADDIM   56
#define D_INNER   1344
#define NHEADS    24
#define CONV_DIM  1600
#define D_IN_PROJ 2968
#define BSZ       2
#define SEQ       2048
#define NTOK      (BSZ*SEQ)
#define EPS       1e-5f

typedef float v2f __attribute__((ext_vector_type(2)));
typedef float v8f __attribute__((ext_vector_type(8)));

// ---------------- RMSNorm over last dim (one block per token) ----------------
__global__ void rmsnorm_kernel(const float* __restrict__ x, const float* __restrict__ w,
                               float* __restrict__ out, int dim) {
  __shared__ float red[256];
  const int tok = blockIdx.x;
  const float* xp = x + (size_t)tok * dim;
  float ss = 0.f;
  for (int i = threadIdx.x; i < dim; i += 256) { float v = xp[i]; ss += v * v; }
  red[threadIdx.x] = ss;
  __syncthreads();
  for (int s = 128; s > 0; s >>= 1) {
    if (threadIdx.x < s) red[threadIdx.x] += red[threadIdx.x + s];
    __syncthreads();
  }
  const float scale = rsqrtf(red[0] / (float)dim + EPS);
  for (int i = threadIdx.x; i < dim; i += 256)
    out[(size_t)tok * dim + i] = w[i] * xp[i] * scale;
}

// ---------------- fp32 WMMA GEMM helpers ------------------------------------
// One stage = 2 K-steps (K advances by 8): 10 b64 loads, 8 WMMAs.
__device__ __forceinline__ void load_stage(const float* __restrict__ Arow,
                                           const float* __restrict__ W0,
                                           const float* __restrict__ W1,
                                           const float* __restrict__ W2,
                                           const float* __restrict__ W3,
                                           int ko_base,
                                           v2f a[2], v2f b0[2], v2f b1[2],
                                           v2f b2[2], v2f b3[2]) {
#pragma unroll
  for (int j = 0; j < 2; ++j) {
    const int ko = ko_base + 4 * j;
    a[j]  = *(const v2f*)(Arow + ko);
    b0[j] = *(const v2f*)(W0 + ko);
    b1[j] = *(const v2f*)(W1 + ko);
    b2[j] = *(const v2f*)(W2 + ko);
    b3[j] = *(const v2f*)(W3 + ko);
  }
}

__device__ __forceinline__ void mma_stage(v8f& acc0, v8f& acc1, v8f& acc2, v8f& acc3,
                                          const v2f a[2], const v2f b0[2], const v2f b1[2],
                                          const v2f b2[2], const v2f b3[2]) {
#pragma unroll
  for (int j = 0; j < 2; ++j) {
    acc0 = __builtin_amdgcn_wmma_f32_16x16x4_f32(false, a[j], false, b0[j], (short)0, acc0, false, false);
    acc1 = __builtin_amdgcn_wmma_f32_16x16x4_f32(false, a[j], false, b1[j], (short)0, acc1, false, false);
    acc2 = __builtin_amdgcn_wmma_f32_16x16x4_f32(false, a[j], false, b2[j], (short)0, acc2, false, false);
    acc3 = __builtin_amdgcn_wmma_f32_16x16x4_f32(false, a[j], false, b3[j], (short)0, acc3, false, false);
  }
}

// ---------------- fp32 WMMA GEMM: C[M,N] = A[M,K] * W[N,K]^T (+resid) -------
// Per wave: one 16(M) x 64(N) tile, 4 accumulators, K in stages of 8 (2
// WMMA-steps), explicitly double-buffered (P/Q) so every compute stage runs
// with the other buffer's 10 loads in flight (distinct live registers ->
// graduated s_wait_loadcnt). Requires K % 16 == 0 (holds: 672, 1344).
// OOB columns: clamp the W row pointer to the last valid row (loads stay in
// bounds; garbage flows only into columns the guarded epilogue never stores).
__global__ void gemm_wmma_kernel(const float* __restrict__ A, const float* __restrict__ W,
                                 const float* __restrict__ resid, float* __restrict__ C,
                                 int M, int N, int K) {
  const int lane = threadIdx.x & 31;
  const int wave = threadIdx.x >> 5;
  const int m0 = blockIdx.y * 16;
  const int nbase = blockIdx.x * 512 + wave * 64;
  if (nbase >= N) return;  // uniform per wave

  const int halfsel = (lane < 16) ? 0 : 2;
  const float* Arow = A + (size_t)(m0 + (lane & 15)) * K;

  const int n0 = nbase + (lane & 15);
  const int n1 = n0 + 16, n2 = n0 + 32, n3 = n0 + 48;
  const int c0 = (n0 < N) ? n0 : (N - 1);
  const int c1 = (n1 < N) ? n1 : (N - 1);
  const int c2 = (n2 < N) ? n2 : (N - 1);
  const int c3 = (n3 < N) ? n3 : (N - 1);
  const float* W0 = W + (size_t)c0 * K;
  const float* W1 = W + (size_t)c1 * K;
  const float* W2 = W + (size_t)c2 * K;
  const float* W3 = W + (size_t)c3 * K;

  v8f acc0 = {}, acc1 = {}, acc2 = {}, acc3 = {};

  v2f aP[2], b0P[2], b1P[2], b2P[2], b3P[2];
  v2f aQ[2], b0Q[2], b1Q[2], b2Q[2], b3Q[2];

  // preload stages 0 and 1
  load_stage(Arow, W0, W1, W2, W3, halfsel,     aP, b0P, b1P, b2P, b3P);
  load_stage(Arow, W0, W1, W2, W3, 8 + halfsel, aQ, b0Q, b1Q, b2Q, b3Q);

  for (int k = 16; k < K; k += 16) {
    mma_stage(acc0, acc1, acc2, acc3, aP, b0P, b1P, b2P, b3P);
    load_stage(Arow, W0, W1, W2, W3, k + halfsel,     aP, b0P, b1P, b2P, b3P);
    mma_stage(acc0, acc1, acc2, acc3, aQ, b0Q, b1Q, b2Q, b3Q);
    load_stage(Arow, W0, W1, W2, W3, k + 8 + halfsel, aQ, b0Q, b1Q, b2Q, b3Q);
  }
  mma_stage(acc0, acc1, acc2, acc3, aP, b0P, b1P, b2P, b3P);
  mma_stage(acc0, acc1, acc2, acc3, aQ, b0Q, b1Q, b2Q, b3Q);

  const int rbase = m0 + ((lane < 16) ? 0 : 8);
#pragma unroll
  for (int r = 0; r < 8; ++r) {
    const size_t off = (size_t)(rbase + r) * N;
    if (n0 < N) { float v = acc0[r]; if (resid) v += resid[off + n0]; C[off + n0] = v; }
    if (n1 < N) { float v = acc1[r]; if (resid) v += resid[off + n1]; C[off + n1] = v; }
    if (n2 < N) { float v = acc2[r]; if (resid) v += resid[off + n2]; C[off + n2] = v; }
    if (n3 < N) { float v = acc3[r]; if (resid) v += resid[off + n3]; C[off + n3] = v; }
  }
}

// ---------------- causal depthwise conv (width 4) + bias + SiLU -------------
__global__ void conv_silu_kernel(const float* __restrict__ zx, const float* __restrict__ cw,
                                 const float* __restrict__ cb, float* __restrict__ xbc) {
  const int idx = blockIdx.x * blockDim.x + threadIdx.x;
  if (idx >= NTOK * CONV_DIM) return;
  const int c = idx % CONV_DIM;
  const int tok = idx / CONV_DIM;
  const int t = tok % SEQ;
  float acc = cb[c];
#pragma unroll
  for (int k = 0; k < D_CONV; ++k) {
    const int tt = t + k - (D_CONV - 1);
    if (tt >= 0)
      acc += zx[(size_t)(tok + k - (D_CONV - 1)) * D_IN_PROJ + D_INNER + c] * cw[c * D_CONV + k];
  }
  acc = acc / (1.f + __expf(-acc));  // SiLU
  xbc[(size_t)tok * CONV_DIM + c] = acc;
}

// ---------------- selective scan: one block per (batch, head) --------------
// thread p (0..55) owns state[p][0..127] in registers; B/C staged via LDS.
__global__ void scan_kernel(const float* __restrict__ zx, const float* __restrict__ xbc,
                            const float* __restrict__ dt_bias, const float* __restrict__ A_log,
                            const float* __restrict__ Dvec, float* __restrict__ yout) {
  const int b = blockIdx.x / NHEADS;
  const int h = blockIdx.x % NHEADS;
  const int tid = threadIdx.x;  // 0..63

  __shared__ float sB[D_STATE];
  __shared__ float sC[D_STATE];

  float state[D_STATE];
#pragma unroll
  for (int n = 0; n < D_STATE; ++n) state[n] = 0.f;

  const float negA = -__expf(A_log[h]);
  const float dtb  = dt_bias[h];
  const float Dh   = Dvec[h];

  for (int t = 0; t < SEQ; ++t) {
    const size_t tok = (size_t)b * SEQ + t;
    const float* xrow = xbc + tok * CONV_DIM;
    __syncthreads();
    sB[tid]      = xrow[D_INNER + tid];
    sB[tid + 64] = xrow[D_INNER + tid + 64];
    sC[tid]      = xrow[D_INNER + D_STATE + tid];
    sC[tid + 64] = xrow[D_INNER + D_STATE + tid + 64];
    __syncthreads();
    if (tid < HEADDIM) {
      const float dtr = zx[tok * D_IN_PROJ + (D_IN_PROJ - NHEADS) + h] + dtb;
      const float dt  = (dtr > 20.f) ? dtr : log1pf(__expf(dtr));  // softplus
      const float dA  = __expf(dt * negA);
      const float xp  = xrow[h * HEADDIM + tid];
      const float coef = dt * xp;
      float y = 0.f;
#pragma unroll
      for (int n = 0; n < D_STATE; ++n) {
        state[n] = state[n] * dA + coef * sB[n];
        y += state[n] * sC[n];
      }
      yout[tok * D_INNER + h * HEADDIM + tid] = y + Dh * xp;
    }
  }
}

// ---------------- gated RMSNorm: out = rmsnorm(y * silu(z)) * gw -----------
__global__ void gatednorm_kernel(const float* __restrict__ y, const float* __restrict__ zx,
                                 const float* __restrict__ gw, float* __restrict__ out) {
  __shared__ float g[D_INNER];
  __shared__ float red[256];
  const int tok = blockIdx.x;
  const int tid = threadIdx.x;
  const float* yp = y + (size_t)tok * D_INNER;
  const float* zp = zx + (size_t)tok * D_IN_PROJ;  // z is the first D_INNER slice
  float ss = 0.f;
  for (int i = tid; i < D_INNER; i += 256) {
    const float z = zp[i];
    const float v = yp[i] * (z / (1.f + __expf(-z)));
    g[i] = v;
    ss += v * v;
  }
  red[tid] = ss;
  __syncthreads();
  for (int s = 128; s > 0; s >>= 1) {
    if (tid < s) red[tid] += red[tid + s];
    __syncthreads();
  }
  const float scale = rsqrtf(red[0] / (float)D_INNER + EPS);
  for (int i = tid; i < D_INNER; i += 256)
    out[(size_t)tok * D_INNER + i] = gw[i] * g[i] * scale;
}

// ---------------------------------------------------------------------------
extern "C" void kernel_launch(void* const* d_in, const int* in_sizes, int n_in,
                              void* d_out, int out_size, void* d_ws, size_t ws_size,
                              hipStream_t stream) {
  const float* hidden  = (const float*)d_in[0];
  const float* norm_w  = (const float*)d_in[1];
  const float* W_in    = (const float*)d_in[2];
  const float* conv_w  = (const float*)d_in[3];
  const float* conv_b  = (const float*)d_in[4];
  const float* dt_bias = (const float*)d_in[5];
  const float* A_log   = (const float*)d_in[6];
  const float* Dv      = (const float*)d_in[7];
  const float* gnorm_w = (const float*)d_in[8];
  const float* W_out   = (const float*)d_in[9];
  float* out = (float*)d_out;

  float* ws    = (float*)d_ws;
  float* u     = ws;                                   // [NTOK, 672]
  float* zx    = u    + (size_t)NTOK * D_MODEL;        // [NTOK, 2968]
  float* xbc   = zx   + (size_t)NTOK * D_IN_PROJ;      // [NTOK, 1600]
  float* yfull = xbc  + (size_t)NTOK * CONV_DIM;       // [NTOK, 1344]
  float* ynorm = xbc;  // reuse: xbc no longer needed after scan completes

  // 1) input RMSNorm
  rmsnorm_kernel<<<NTOK, 256, 0, stream>>>(hidden, norm_w, u, D_MODEL);

  // 2) in-projection GEMM (WMMA f32): zx = u @ W_in^T
  {
    dim3 grid((D_IN_PROJ + 511) / 512, NTOK / 16);
    gemm_wmma_kernel<<<grid, 256, 0, stream>>>(u, W_in, nullptr, zx,
                                               NTOK, D_IN_PROJ, D_MODEL);
  }

  // 3) depthwise causal conv + SiLU on xBC slice
  {
    const int n = NTOK * CONV_DIM;
    conv_silu_kernel<<<(n + 255) / 256, 256, 0, stream>>>(zx, conv_w, conv_b, xbc);
  }

  // 4) sequential selective-state scan
  scan_kernel<<<BSZ * NHEADS, 64, 0, stream>>>(zx, xbc, dt_bias, A_log, Dv, yfull);

  // 5) gated RMSNorm
  gatednorm_kernel<<<NTOK, 256, 0, stream>>>(yfull, zx, gnorm_w, ynorm);

  // 6) out-projection GEMM (WMMA f32) + residual: out = hidden + ynorm @ W_out^T
  {
    dim3 grid((D_MODEL + 511) / 512, NTOK / 16);
    gemm_wmma_kernel<<<grid, 256, 0, stream>>>(ynorm, W_out, hidden, out,
                                               NTOK, D_MODEL, D_INNER);
  }
}